// ColorNorm_6536940224723
// MI455X (gfx1250) — compile-verified
//
#include <hip/hip_runtime.h>

typedef __attribute__((ext_vector_type(2))) float v2f;
typedef __attribute__((ext_vector_type(8))) float v8f;

#define TCOLS   256     // spatial columns staged per tile
#define RSTRIDE 260     // padded LDS row stride (dwords), even + multiple of 4
#define WPB     8       // waves per block (256 threads)
#define WGS     16      // workgroups per batch for the stats pass

// ---------------------------------------------------------------------------
// Kernel 0: zero the stats accumulation buffer (7x7 Gram block per batch).
// ---------------------------------------------------------------------------
__global__ void zero_kernel(float* __restrict__ p, int n) {
    int i = blockIdx.x * blockDim.x + threadIdx.x;
    if (i < n) p[i] = 0.0f;
}

// ---------------------------------------------------------------------------
// Kernel 1: Gram-matrix statistics via V_WMMA_F32_16X16X4_F32.
// G is 7 x N per batch: rows 0-2 = src channels, 3-5 = dst channels, 6 = ones.
// D = G * G^T accumulated K=4 columns at a time in the WMMA accumulator.
// stats[b][r*7+j] += D[r][j] for r,j in 0..6 via global f32 atomics.
// ---------------------------------------------------------------------------
__global__ void __launch_bounds__(WPB * 32)
stats_kernel(const float* __restrict__ src, const float* __restrict__ dst,
             float* __restrict__ stats, int n) {
    __shared__ __align__(16) float lds[WPB * 6 * RSTRIDE];
    __shared__ __align__(16) float onesrow[RSTRIDE];
    __shared__ __align__(16) float zrow[RSTRIDE];

    const int tid  = threadIdx.x;
    const int w    = tid >> 5;        // wave id in block
    const int lane = tid & 31;
    const int b    = blockIdx.y;      // batch

    // Shared constant rows (row 6 = ones, rows 7..15 = zeros), filled by wave 0.
    if (w == 0) {
        for (int cc = lane; cc < RSTRIDE; cc += 32) {
            onesrow[cc] = 1.0f;
            zrow[cc]    = 0.0f;
        }
    }
    __syncthreads();

    float* my = lds + w * (6 * RSTRIDE);

    // WMMA fragment addressing: lane L holds matrix row m = L&15,
    // K-pair k01 = (L>>4)*2 (per 32-bit 16x4 A layout; B fragment is the
    // lane-wise transpose and therefore identical per lane).
    const int m   = lane & 15;
    const int k01 = (lane >> 4) << 1;
    const float* frag_base;
    if (m < 6)       frag_base = my + m * RSTRIDE + k01;
    else if (m == 6) frag_base = onesrow + k01;
    else             frag_base = zrow + k01;

    v8f acc = {0.0f, 0.0f, 0.0f, 0.0f, 0.0f, 0.0f, 0.0f, 0.0f};

    const int num_tiles       = n / TCOLS;
    const int waves_per_batch = gridDim.x * WPB;
    const size_t cbase        = (size_t)b * 3 * (size_t)n;

    for (int tile = blockIdx.x * WPB + w; tile < num_tiles; tile += waves_per_batch) {
        const int col0 = tile * TCOLS;

        // Stage 6 rows x 256 cols into LDS, fully coalesced b128 traffic.
#pragma unroll
        for (int r = 0; r < 3; ++r) {
            const float* gs = src + cbase + (size_t)r * n + col0;
            const float* gd = dst + cbase + (size_t)r * n + col0;
            *(float4*)(my + r * RSTRIDE + lane * 4)             = *(const float4*)(gs + lane * 4);
            *(float4*)(my + r * RSTRIDE + 128 + lane * 4)       = *(const float4*)(gs + 128 + lane * 4);
            *(float4*)(my + (3 + r) * RSTRIDE + lane * 4)       = *(const float4*)(gd + lane * 4);
            *(float4*)(my + (3 + r) * RSTRIDE + 128 + lane * 4) = *(const float4*)(gd + 128 + lane * 4);
        }

        // 64 WMMA steps of K=4 cover the 256 staged columns.
#pragma unroll 4
        for (int g = 0; g < TCOLS / 4; ++g) {
            v2f ab = *(const v2f*)(frag_base + g * 4);   // ds_load_b64
            acc = __builtin_amdgcn_wmma_f32_16x16x4_f32(
                false, ab, false, ab, (short)0, acc, false, false);
        }
    }

    // D layout: VGPR r holds row M=r for lanes 0-15 (col N = lane).
    // We need rows 0..6, cols 0..6.
    if (lane < 7) {
#pragma unroll
        for (int r = 0; r < 7; ++r)
            atomicAdd(&stats[b * 49 + r * 7 + lane], acc[r]);
    }
}

// ---------------------------------------------------------------------------
// Kernel 2: per-batch 3x3 solve (fp64). One lane per batch.
// stats(i,j)=sum A_i A_j ; stats(3+i,j)=sum B_i A_j ; stats(6,i)=sum A_i ;
// stats(6,3+i)=sum B_i ; stats(6,6)=N.
// Writes coef[b][0..8]=x (row-major), coef[b][9..11]=t.
// ---------------------------------------------------------------------------
__global__ void solve_kernel(const float* __restrict__ stats,
                             float* __restrict__ coef, int nbatch) {
    int b = blockIdx.x * blockDim.x + threadIdx.x;
    if (b >= nbatch) return;
    const float* s = stats + b * 49;
#define S(r, j) ((double)s[(r) * 7 + (j)])

    double cnt = S(6, 6);
    double inv_cnt = 1.0 / cnt;
    double sa[3], sb[3], am[3], bm[3];
#pragma unroll
    for (int i = 0; i < 3; ++i) {
        sa[i] = S(6, i);
        sb[i] = S(6, 3 + i);
        am[i] = sa[i] * inv_cnt;
        bm[i] = sb[i] * inv_cnt;
    }

    double aat[3][3], bat[3][3];
#pragma unroll
    for (int i = 0; i < 3; ++i)
#pragma unroll
        for (int j = 0; j < 3; ++j) {
            aat[i][j] = S(i, j) - sa[i] * sa[j] * inv_cnt + ((i == j) ? 0.001 : 0.0);
            bat[i][j] = S(3 + i, j) - sb[i] * sa[j] * inv_cnt;
        }
#undef S

    // 3x3 inverse via adjugate.
    double a00 = aat[0][0], a01 = aat[0][1], a02 = aat[0][2];
    double a10 = aat[1][0], a11 = aat[1][1], a12 = aat[1][2];
    double a20 = aat[2][0], a21 = aat[2][1], a22 = aat[2][2];
    double c00 =  (a11 * a22 - a12 * a21);
    double c01 = -(a10 * a22 - a12 * a20);
    double c02 =  (a10 * a21 - a11 * a20);
    double c10 = -(a01 * a22 - a02 * a21);
    double c11 =  (a00 * a22 - a02 * a20);
    double c12 = -(a00 * a21 - a01 * a20);
    double c20 =  (a01 * a12 - a02 * a11);
    double c21 = -(a00 * a12 - a02 * a10);
    double c22 =  (a00 * a11 - a01 * a10);
    double id  = 1.0 / (a00 * c00 + a01 * c01 + a02 * c02);
    double inv[3][3];
    inv[0][0] = c00 * id; inv[0][1] = c10 * id; inv[0][2] = c20 * id;
    inv[1][0] = c01 * id; inv[1][1] = c11 * id; inv[1][2] = c21 * id;
    inv[2][0] = c02 * id; inv[2][1] = c12 * id; inv[2][2] = c22 * id;

    double x[3][3], t[3];
#pragma unroll
    for (int i = 0; i < 3; ++i) {
#pragma unroll
        for (int j = 0; j < 3; ++j) {
            double v = 0.0;
#pragma unroll
            for (int k = 0; k < 3; ++k) v += bat[i][k] * inv[k][j];
            x[i][j] = v;
        }
        t[i] = bm[i] - (x[i][0] * am[0] + x[i][1] * am[1] + x[i][2] * am[2]);
    }

#pragma unroll
    for (int i = 0; i < 3; ++i) {
#pragma unroll
        for (int j = 0; j < 3; ++j) coef[b * 12 + i * 3 + j] = (float)x[i][j];
        coef[b * 12 + 9 + i] = (float)t[i];
    }
}

// ---------------------------------------------------------------------------
// Kernel 3: apply out[i][n] = sum_j x[i][j]*src[j][n] + t[i], float4 streams.
// ---------------------------------------------------------------------------
__global__ void __launch_bounds__(256)
apply_kernel(const float* __restrict__ src, const float* __restrict__ coef,
             float* __restrict__ out, int n) {
    __shared__ float c[12];
    const int tid = threadIdx.x;
    const int b   = blockIdx.y;
    if (tid < 12) c[tid] = coef[b * 12 + tid];
    __syncthreads();

    const long long n4 = (long long)(n >> 2);
    const long long i4 = (long long)blockIdx.x * blockDim.x + tid;
    if (i4 >= n4) return;

    const float4* s = (const float4*)(src + (size_t)b * 3 * (size_t)n);
    float4*       o = (float4*)(out + (size_t)b * 3 * (size_t)n);

    float4 a0 = s[i4];
    float4 a1 = s[n4 + i4];
    float4 a2 = s[2 * n4 + i4];

#pragma unroll
    for (int i = 0; i < 3; ++i) {
        const float m0 = c[i * 3 + 0], m1 = c[i * 3 + 1], m2 = c[i * 3 + 2], tt = c[9 + i];
        float4 r;
        r.x = m0 * a0.x + m1 * a1.x + m2 * a2.x + tt;
        r.y = m0 * a0.y + m1 * a1.y + m2 * a2.y + tt;
        r.z = m0 * a0.z + m1 * a1.z + m2 * a2.z + tt;
        r.w = m0 * a0.w + m1 * a1.w + m2 * a2.w + tt;
        o[(long long)i * n4 + i4] = r;
    }
}

// ---------------------------------------------------------------------------
extern "C" void kernel_launch(void* const* d_in, const int* in_sizes, int n_in,
                              void* d_out, int out_size, void* d_ws, size_t ws_size,
                              hipStream_t stream) {
    const float* src = (const float*)d_in[0];
    const float* dst = (const float*)d_in[1];
    float* out = (float*)d_out;

    const int B = 16, C = 3;
    const int n = in_sizes[0] / (B * C);   // 1024*1024 spatial elements

    float* stats = (float*)d_ws;                       // B*49 floats
    float* coef  = (float*)((char*)d_ws + 4096);       // B*12 floats

    // Pass 0: zero stats accumulators.
    zero_kernel<<<dim3((B * 49 + 255) / 256), dim3(256), 0, stream>>>(stats, B * 49);

    // Pass 1: Gram statistics with f32 WMMA.
    stats_kernel<<<dim3(WGS, B), dim3(WPB * 32), 0, stream>>>(src, dst, stats, n);

    // Pass 2: tiny per-batch 3x3 solve.
    solve_kernel<<<dim3(1), dim3(32), 0, stream>>>(stats, coef, B);

    // Pass 3: streaming affine apply.
    const int n4 = n / 4;
    apply_kernel<<<dim3((n4 + 255) / 256, B), dim3(256), 0, stream>>>(src, coef, out, n);
}